// MFM_70875550319036
// MI455X (gfx1250) — compile-verified
//
#include <hip/hip_runtime.h>

// Max-Feature-Map over adjacent channel pairs.
// x: (32, 128, 112, 112) fp32  ->  y: (32, 64, 112, 112) fp32
//
// Pure HBM-streaming op: ~308 MB traffic, floor ~13.2 us at 23.3 TB/s.
// 128-bit coalesced NT loads/stores, 4-way unroll (8 loads in flight per
// thread), wave32-friendly 256-thread blocks. Fast path exploits
// stride4 % HW4 == 0 so the plane-offset modulo is computed once.

typedef __attribute__((ext_vector_type(4))) float v4f;

#define MFM_HW   12544   // 112*112
#define MFM_HW4  3136    // HW / 4 (float4 units)

__device__ __forceinline__ v4f mfm_vmax4(v4f a, v4f b) {
    v4f m;
    m.x = __builtin_fmaxf(a.x, b.x);
    m.y = __builtin_fmaxf(a.y, b.y);
    m.z = __builtin_fmaxf(a.z, b.z);
    m.w = __builtin_fmaxf(a.w, b.w);
    return m;
}

// Fast path preconditions (checked on host):
//   n4 == 4 * stride4, stride4 == grid * 256, stride4 % MFM_HW4 == 0.
// The last one means hw4 = i % HW4 is identical for all 4 unrolled elements:
// one modulo, then pure adds. All 8 NT loads issue before any wait.
__global__ __launch_bounds__(256)
void mfm_exact_kernel(const float* __restrict__ x, float* __restrict__ y,
                      int stride4) {
    const int i0 = blockIdx.x * 256 + threadIdx.x;  // output float4 index, k=0
    const v4f* __restrict__ x4 = (const v4f*)x;
    v4f* __restrict__ y4 = (v4f*)y;

    const int hw4 = i0 % MFM_HW4;      // same for all k (stride4 % HW4 == 0)
    const int in0 = 2 * i0 - hw4;      // float4 index of channel 2c, k=0
    const int instep = 2 * stride4;    // input float4 step between unrolls

    v4f a[4], b[4];
#pragma unroll
    for (int k = 0; k < 4; ++k) {
        const int ik = in0 + k * instep;
        a[k] = __builtin_nontemporal_load(x4 + ik);            // ch 2c
        b[k] = __builtin_nontemporal_load(x4 + ik + MFM_HW4);  // ch 2c+1
    }
#pragma unroll
    for (int k = 0; k < 4; ++k) {
        __builtin_nontemporal_store(mfm_vmax4(a[k], b[k]),
                                    y4 + i0 + k * stride4);
    }
}

// Generic guarded fallback (grid-stride), used only if sizes don't tile.
__global__ __launch_bounds__(256)
void mfm_guard_kernel(const float* __restrict__ x, float* __restrict__ y,
                      int n4, int stride4) {
    const v4f* __restrict__ x4 = (const v4f*)x;
    v4f* __restrict__ y4 = (v4f*)y;
    for (int i = blockIdx.x * 256 + threadIdx.x; i < n4; i += stride4) {
        const int hw4 = i % MFM_HW4;
        const int in0 = 2 * i - hw4;
        v4f a = __builtin_nontemporal_load(x4 + in0);
        v4f b = __builtin_nontemporal_load(x4 + in0 + MFM_HW4);
        __builtin_nontemporal_store(mfm_vmax4(a, b), y4 + i);
    }
}

extern "C" void kernel_launch(void* const* d_in, const int* in_sizes, int n_in,
                              void* d_out, int out_size, void* d_ws, size_t ws_size,
                              hipStream_t stream) {
    (void)n_in; (void)out_size; (void)d_ws; (void)ws_size;

    const float* x = (const float*)d_in[0];
    float* y = (float*)d_out;

    const long long n_in_elems = (long long)in_sizes[0];  // 32*128*112*112
    const int n4 = (int)(n_in_elems / 8);                 // output float4 count

    constexpr int BLOCK = 256;   // 8 wave32s per block
    constexpr int UNROLL = 4;

    const int stride4 = n4 / UNROLL;
    if (n4 % (BLOCK * UNROLL) == 0 && stride4 % MFM_HW4 == 0) {
        const int grid = stride4 / BLOCK;        // 6,272 blocks
        mfm_exact_kernel<<<grid, BLOCK, 0, stream>>>(x, y, stride4);
    } else {
        const int grid = (n4 + BLOCK * UNROLL - 1) / (BLOCK * UNROLL);
        mfm_guard_kernel<<<grid, BLOCK, 0, stream>>>(x, y, n4, grid * BLOCK);
    }
}